// DS2_87582973100612
// MI455X (gfx1250) — compile-verified
//
#include <hip/hip_runtime.h>

// ---------------------------------------------------------------------------
// out[p, d*2+c] = x[p, d] * u[d, c],  u = beta^2 / rowsum(beta^2)  (all f32)
//
// Pure HBM-bound op (300 MiB / launch, ~13.5 us floor @ 23.3 TB/s).
// Implemented as Out = X * B with B the diagonal-expanded 32x64 u-matrix,
// tiled onto V_WMMA_F32_16X16X4_F32 (exact in f32), with LDS repack so the
// 200 MiB output stream goes out as 128-bit NT coalesced stores.
// ---------------------------------------------------------------------------

typedef __attribute__((ext_vector_type(2))) float v2f;
typedef __attribute__((ext_vector_type(4))) float v4f;
typedef __attribute__((ext_vector_type(8))) float v8f;

#define D_DIM      32
#define DC_DIM     64                 // D*C
#define PIX_TOTAL  (16 * 160 * 320)   // 819200 pixels
#define GROUPS     (PIX_TOTAL / 16)   // 51200 groups of 16 pixels
#define ROWSTRIDE  68                 // 64 + 4 pad floats -> conflict-free LDS writes
#define NBLOCKS    800                // 800*8 waves * 8 groups = 51200 exactly

__global__ __launch_bounds__(256) void
mass_prototype_wmma(const float* __restrict__ in,
                    const float* __restrict__ beta,
                    float* __restrict__ out)
{
    __shared__ float lds[8 * 16 * ROWSTRIDE];   // 34.8 KB, private region per wave

    const int tid  = threadIdx.x;
    const int lane = tid & 31;
    const int wave = tid >> 5;
    const int col  = lane & 15;   // N column / A row (M)
    const int h    = lane >> 4;   // lane half: owns K rows {2h, 2h+1}
    float* wlds = &lds[wave * 16 * ROWSTRIDE];

    // ---- Build wave-constant sparse B tiles in VGPRs ------------------------
    // Output tile t covers dc columns [16t,16t+16) i.e. d in [8t,8t+8), split
    // into K-slices s=0,1 (4 d's each). B_s[k, n] = u[8t+4s+k, n&1] iff
    // n>>1 == 4s+k, else 0.  VGPR layout (ISA 7.12.2): lane half h, comp j -> K=2h+j.
    const int dloc = col >> 1;        // which local d (0..7) owns column `col`
    const int cch  = col & 1;         // channel c
    const int s_m  = dloc >> 2;       // slice holding the nonzero
    const int k_m  = dloc & 3;        // K row holding the nonzero
    const bool mine = ((k_m >> 1) == h);

    v2f Bm[4][2];
    #pragma unroll
    for (int t = 0; t < 4; ++t) {
        const int d = 8 * t + dloc;
        const v2f bb = *(const v2f*)&beta[2 * d];       // beta[d,0], beta[d,1]
        const float bc = cch ? bb.y : bb.x;
        const float u  = (bc * bc) / (bb.x * bb.x + bb.y * bb.y);
        #pragma unroll
        for (int s = 0; s < 2; ++s)
            #pragma unroll
            for (int j = 0; j < 2; ++j)   // constant indices -> stays in VGPRs
                Bm[t][s][j] = (mine && (s == s_m) && (j == k_m % 2)) ? u : 0.0f;
    }

    // ---- Grid-stride over 16-pixel groups (uniform trip count: 8/wave) -----
    const int wavesTotal = gridDim.x * (blockDim.x >> 5);
    const int waveId     = blockIdx.x * (blockDim.x >> 5) + wave;

    for (int g = waveId; g < GROUPS; g += wavesTotal) {
        const long pixBase = (long)g * 16;
        const float* src = in + pixBase * D_DIM;

        // prefetch next group's input stream
        __builtin_prefetch(src + (long)wavesTotal * 16 * D_DIM, 0, 1);

        #pragma unroll
        for (int t = 0; t < 4; ++t) {
            // A slice for K-slice s: lane(h,col) needs x[pixel=col, 8t+4s+2h .. +1]
            // -> single b64 load per lane, exactly matching the f32 A layout.
            const v2f A0 = *(const v2f*)(src + (long)col * D_DIM + (8 * t     + 2 * h));
            const v2f A1 = *(const v2f*)(src + (long)col * D_DIM + (8 * t + 4 + 2 * h));

            v8f acc = {0.f, 0.f, 0.f, 0.f, 0.f, 0.f, 0.f, 0.f};
            acc = __builtin_amdgcn_wmma_f32_16x16x4_f32(
                      false, A0, false, Bm[t][0], (short)0, acc, false, false);
            acc = __builtin_amdgcn_wmma_f32_16x16x4_f32(
                      false, A1, false, Bm[t][1], (short)0, acc, false, false);

            // D layout: VGPR r, lane -> (M = r + 8h, N = col). Park in LDS.
            #pragma unroll
            for (int r = 0; r < 8; ++r)
                wlds[(r + 8 * h) * ROWSTRIDE + 16 * t + col] = acc[r];
        }

        __syncthreads();   // make cross-lane LDS writes visible (uniform count)

        // Repack: 8 x (32 lanes x 16B) = fully coalesced 512B NT stores.
        float* dst = out + pixBase * DC_DIM;
        #pragma unroll
        for (int q = 0; q < 8; ++q) {
            const int p = 2 * q + h;
            const v4f v = *(const v4f*)&wlds[p * ROWSTRIDE + col * 4];
            __builtin_nontemporal_store(v, (v4f*)(dst + (long)p * DC_DIM + col * 4));
        }

        __syncthreads();   // WAR: reads done before next iteration overwrites
    }
}

extern "C" void kernel_launch(void* const* d_in, const int* in_sizes, int n_in,
                              void* d_out, int out_size, void* d_ws, size_t ws_size,
                              hipStream_t stream) {
    const float* in   = (const float*)d_in[0];
    const float* beta = (const float*)d_in[1];
    float* out        = (float*)d_out;
    hipLaunchKernelGGL(mass_prototype_wmma, dim3(NBLOCKS), dim3(256), 0, stream,
                       in, beta, out);
}